// DGLLayer_31482110279708
// MI455X (gfx1250) — compile-verified
//
#include <hip/hip_runtime.h>

typedef float v2f __attribute__((ext_vector_type(2)));
typedef float v8f __attribute__((ext_vector_type(8)));

#define L_DIM 1024
#define N_DIM 8
#define C_DIM 2048
#define O_DIM 2048
#define K_DIM 1024
#define NC (N_DIM * C_DIM)   // 16384
#define NO (N_DIM * O_DIM)   // 16384

// ---------------------------------------------------------------------------
// 1) x_avg[n][c] = mean over L of x[l,n,c].  x layout: l*NC + n*C + c.
// ---------------------------------------------------------------------------
__global__ void mean_over_L(const float* __restrict__ x, float* __restrict__ x_avg) {
    int t = blockIdx.x * 256 + threadIdx.x;      // t = n*C + c, 0..16383
    float s = 0.0f;
    for (int l = 0; l < L_DIM; ++l) s += x[(size_t)l * NC + t];
    x_avg[t] = s * (1.0f / (float)L_DIM);
}

// ---------------------------------------------------------------------------
// 2) LayerNorm over C per sample n.  One block per n.
// ---------------------------------------------------------------------------
__global__ void layernorm_k(const float* __restrict__ xa, const float* __restrict__ g,
                            const float* __restrict__ b, float* __restrict__ xn) {
    __shared__ float s1[256];
    __shared__ float s2[256];
    int n = blockIdx.x;
    float a = 0.0f, q = 0.0f;
    for (int c = threadIdx.x; c < C_DIM; c += 256) {
        float v = xa[n * C_DIM + c];
        a += v;
        q += v * v;
    }
    s1[threadIdx.x] = a; s2[threadIdx.x] = q;
    __syncthreads();
    for (int st = 128; st > 0; st >>= 1) {
        if (threadIdx.x < st) {
            s1[threadIdx.x] += s1[threadIdx.x + st];
            s2[threadIdx.x] += s2[threadIdx.x + st];
        }
        __syncthreads();
    }
    float mu  = s1[0] * (1.0f / (float)C_DIM);
    float var = s2[0] * (1.0f / (float)C_DIM) - mu * mu;
    float rs  = rsqrtf(var + 1e-5f);
    for (int c = threadIdx.x; c < C_DIM; c += 256) {
        float v = (xa[n * C_DIM + c] - mu) * rs;
        xn[n * C_DIM + c] = v * g[c] + b[c];
    }
}

// ---------------------------------------------------------------------------
// 3) Skinny GEMM: out[n][o] = relu?( in[n][:] . W[o][:] + bias[o] ), n=0..7.
//    One thread per output o; streams W once (bandwidth roof).
// ---------------------------------------------------------------------------
__global__ void gemm_m8(const float* __restrict__ in, const float* __restrict__ W,
                        const float* __restrict__ bias, float* __restrict__ out,
                        int do_relu) {
    int o = blockIdx.x * 256 + threadIdx.x;      // 0..2047
    const float* wr = W + (size_t)o * C_DIM;
    float acc[N_DIM] = {};
    for (int c = 0; c < C_DIM; ++c) {
        float w = wr[c];
#pragma unroll
        for (int nn = 0; nn < N_DIM; ++nn)
            acc[nn] = fmaf(w, in[nn * C_DIM + c], acc[nn]);
    }
    float bv = bias[o];
#pragma unroll
    for (int nn = 0; nn < N_DIM; ++nn) {
        float r = acc[nn] + bv;
        if (do_relu) r = fmaxf(r, 0.0f);
        out[nn * O_DIM + o] = r;
    }
}

// ---------------------------------------------------------------------------
// 4) Exact top-k by rank counting (matches jax.lax.top_k ordering).
// ---------------------------------------------------------------------------
__global__ void topk_k(const float* __restrict__ logits, int* __restrict__ idx_out) {
    __shared__ float v[O_DIM];
    int n = blockIdx.x;
    for (int i = threadIdx.x; i < O_DIM; i += 256) v[i] = logits[n * O_DIM + i];
    __syncthreads();
    for (int i = threadIdx.x; i < O_DIM; i += 256) {
        float my = v[i];
        int rank = 0;
        for (int j = 0; j < O_DIM; ++j) {
            float o = v[j];
            rank += (o > my) || (o == my && j < i);
        }
        if (rank < K_DIM) idx_out[n * K_DIM + rank] = i;
    }
}

// ---------------------------------------------------------------------------
// 5) Zero-fill output tensor (unselected rows must be exactly 0).
// ---------------------------------------------------------------------------
__global__ void zero_out(float4* __restrict__ p) {
    float4 z = {0.0f, 0.0f, 0.0f, 0.0f};
    p[(size_t)blockIdx.x * 256 + threadIdx.x] = z;
}

// ---------------------------------------------------------------------------
// 6) Main gathered GEMM + scatter, fp32 WMMA with register tiling.
//    Each wave: 32(k) x 64(l) macro-tile = 2 A-frags x 4 B-frags ->
//    8 f32 16x16 accumulators.  Per C-step of 4: 6 float2 loads feed
//    8 V_WMMA_F32_16X16X4_F32 (0.75 loads/wmma); accumulator rotation
//    gives an 8-deep RAW distance on each acc.
//    A(16x4): lane = m + 16*(K>=2), vgpr = K&1 -> float2 of gathered W row.
//    B(4x16): symmetric; float2 of x column (x[l,n,c] contiguous in c).
//    D(16x16): vgpr v, lanes 0-15 -> (M=v, N=lane), lanes 16-31 -> (M=v+8).
// ---------------------------------------------------------------------------
__global__ void spmm_wmma(const float* __restrict__ x, const float* __restrict__ lw,
                          const float* __restrict__ lb, const int* __restrict__ idx,
                          float* __restrict__ out) {
    const int lane = threadIdx.x & 31;
    const int wave = threadIdx.x >> 5;
    const int n    = blockIdx.y;
    const int tile = blockIdx.x * 8 + wave;      // 0..511
    const int kt   = tile >> 4;                  // 0..31 : 32-row k block
    const int lt   = tile & 15;                  // 0..15 : 64-col l block
    const int m    = lane & 15;
    const int hi   = lane >> 4;
    const int jb   = hi * 2;

    // A fragment sources: 2 sets of 16 gathered weight rows
    const int kbase = n * K_DIM + kt * 32;
    const int r0 = idx[kbase + m];
    const int r1 = idx[kbase + 16 + m];
    const float* wr0 = lw + (size_t)r0 * C_DIM + jb;
    const float* wr1 = lw + (size_t)r1 * C_DIM + jb;

    // B fragment sources: 4 sets of 16 x-columns
    const int lbase = lt * 64;
    const float* xc0 = x + (size_t)(lbase + m) * NC + n * C_DIM + jb;
    const float* xc1 = xc0 + (size_t)16 * NC;
    const float* xc2 = xc0 + (size_t)32 * NC;
    const float* xc3 = xc0 + (size_t)48 * NC;

    v8f a00 = {}, a01 = {}, a02 = {}, a03 = {};
    v8f a10 = {}, a11 = {}, a12 = {}, a13 = {};

    for (int c = 0; c < C_DIM; c += 4) {
        v2f fa0 = *(const v2f*)(wr0 + c);
        v2f fa1 = *(const v2f*)(wr1 + c);
        v2f fb0 = *(const v2f*)(xc0 + c);
        v2f fb1 = *(const v2f*)(xc1 + c);
        v2f fb2 = *(const v2f*)(xc2 + c);
        v2f fb3 = *(const v2f*)(xc3 + c);
        a00 = __builtin_amdgcn_wmma_f32_16x16x4_f32(false, fa0, false, fb0, (short)0, a00, false, false);
        a01 = __builtin_amdgcn_wmma_f32_16x16x4_f32(false, fa0, false, fb1, (short)0, a01, false, false);
        a02 = __builtin_amdgcn_wmma_f32_16x16x4_f32(false, fa0, false, fb2, (short)0, a02, false, false);
        a03 = __builtin_amdgcn_wmma_f32_16x16x4_f32(false, fa0, false, fb3, (short)0, a03, false, false);
        a10 = __builtin_amdgcn_wmma_f32_16x16x4_f32(false, fa1, false, fb0, (short)0, a10, false, false);
        a11 = __builtin_amdgcn_wmma_f32_16x16x4_f32(false, fa1, false, fb1, (short)0, a11, false, false);
        a12 = __builtin_amdgcn_wmma_f32_16x16x4_f32(false, fa1, false, fb2, (short)0, a12, false, false);
        a13 = __builtin_amdgcn_wmma_f32_16x16x4_f32(false, fa1, false, fb3, (short)0, a13, false, false);
    }

    // Scatter: out[l, n, o] = acc + l_b[o], o = idx row, exact positions.
    v8f accf0[4] = {a00, a01, a02, a03};
    v8f accf1[4] = {a10, a11, a12, a13};
#pragma unroll
    for (int f = 0; f < 2; ++f) {
#pragma unroll
        for (int v = 0; v < 8; ++v) {
            int o  = idx[kbase + f * 16 + v + 8 * hi];
            float bv = lb[o];
#pragma unroll
            for (int g = 0; g < 4; ++g) {
                int lcol = lbase + g * 16 + m;
                float val = (f == 0 ? accf0[g][v] : accf1[g][v]) + bv;
                out[(size_t)lcol * NO + n * O_DIM + o] = val;
            }
        }
    }
}

// ---------------------------------------------------------------------------
extern "C" void kernel_launch(void* const* d_in, const int* in_sizes, int n_in,
                              void* d_out, int out_size, void* d_ws, size_t ws_size,
                              hipStream_t stream) {
    (void)in_sizes; (void)n_in; (void)out_size; (void)ws_size;

    const float* x    = (const float*)d_in[0];   // [L,N,C]
    const float* l_w  = (const float*)d_in[1];   // [O,C]
    const float* l_b  = (const float*)d_in[2];   // [O]
    const float* ln_g = (const float*)d_in[3];   // [C]
    const float* ln_b = (const float*)d_in[4];   // [C]
    const float* gw   = (const float*)d_in[5];   // [O,O]
    const float* gb   = (const float*)d_in[6];   // [O]

    float* ws     = (float*)d_ws;
    float* x_avg  = ws;                          // [N,C]
    float* x_norm = ws + NC;                     // [N,C]
    float* gate   = ws + 2 * NC;                 // [N,O]
    float* logits = ws + 3 * NC;                 // [N,O]

    float* out    = (float*)d_out;               // [L,N,O] floats ...
    int*  idx_out = (int*)(out + (size_t)L_DIM * NO);  // ... then [N,K] int32

    mean_over_L<<<NC / 256, 256, 0, stream>>>(x, x_avg);
    layernorm_k<<<N_DIM, 256, 0, stream>>>(x_avg, ln_g, ln_b, x_norm);
    gemm_m8<<<O_DIM / 256, 256, 0, stream>>>(x_norm, l_w, l_b, gate, 0);
    gemm_m8<<<O_DIM / 256, 256, 0, stream>>>(gate, gw, gb, logits, 1);
    topk_k<<<N_DIM, 256, 0, stream>>>(logits, idx_out);
    zero_out<<<(L_DIM * (size_t)NO) / 4 / 256, 256, 0, stream>>>((float4*)out);
    spmm_wmma<<<dim3(64, N_DIM), 256, 0, stream>>>(x, l_w, l_b, idx_out, out);
}